// ISTFT_75866302316578
// MI455X (gfx1250) — compile-verified
//
#include <hip/hip_runtime.h>
#include <hip/hip_bf16.h>
#include <math.h>

// ISTFT as one fused fp32 GEMM: out[8192,600] = X[8192,3084] * W[3084,600]
// W built on-device each call (deterministic) in WMMA B-fragment order.

typedef __attribute__((ext_vector_type(2))) float v2f;
typedef __attribute__((ext_vector_type(8))) float v8f;

#define KTOT   3084            // 257 freqs * 6 frames * 2 (re,im)
#define KSTEPS 771             // KTOT / 4 (WMMA K=4 per step)
#define NTILES 40              // 640 padded output columns / 16
#define FRAG   64              // floats per (kstep, ntile) B fragment: 32 lanes * 2
#define NOUT   600
#define KCH    768             // K-chunk staged in LDS (multiple of 4)
#define LDSTR  772             // padded LDS row stride (772 mod 64 == 4 -> conflict-free)

// ---------------------------------------------------------------------------
// Kernel 1: build W in WMMA B-fragment layout.
// Flat index gid = ((kk*NTILES + nt)*32 + lane)*2 + r
//   lane: N = nt*16 + lane%16 ; K = 4*kk + 2*(lane/16) + r   (mirrors A layout)
// ---------------------------------------------------------------------------
__global__ __launch_bounds__(256)
void ISTFT_buildW(float* __restrict__ Wf) {
  int gid = blockIdx.x * 256 + threadIdx.x;
  if (gid >= KSTEPS * NTILES * FRAG) return;

  int r    = gid & 1;
  int l    = (gid >> 1) & 31;
  int rest = gid >> 6;
  int nt   = rest % NTILES;
  int kk   = rest / NTILES;

  int k   = (kk << 2) + ((l >> 4) << 1) + r;   // 0..3083
  int f   = k / 12;                            // 0..256
  int rem = k - f * 12;
  int tf  = rem >> 1;                          // frame 0..5
  int d   = rem & 1;                           // 0=re, 1=im
  int n   = (nt << 4) + (l & 15);              // output column (padded to 640)

  float v = 0.0f;
  if (n < NOUT) {
    int labs = n + 256;                        // center=True trim of n_fft/2
    int nn   = labs - (tf << 7);               // position inside frame
    if (nn >= 0 && nn < 512) {
      const float c0 = 6.283185307179586f / 512.0f;
      // window-square envelope (COLA denominator) over the 6 frames
      float env = 0.0f;
      #pragma unroll
      for (int tp = 0; tp < 6; ++tp) {
        int nn2 = labs - (tp << 7);
        if (nn2 >= 0 && nn2 < 512) {
          float w = 0.5f * (1.0f - cosf(c0 * (float)nn2));
          env += w * w;
        }
      }
      float wn   = 0.5f * (1.0f - cosf(c0 * (float)nn));   // periodic Hann
      bool  edge = (f == 0) || (f == 256);
      float scl  = (edge ? 1.0f : 2.0f) / 512.0f;          // irfft 1/N + Hermitian 2x
      int   m    = (f * nn) & 511;                         // exact angle reduction
      float ang  = c0 * (float)m;
      float basis = d ? -sinf(ang) : cosf(ang);
      if (d && edge) basis = 0.0f;                         // irfft drops imag(DC/Nyq)
      v = scl * wn / env * basis;
    }
  }
  Wf[gid] = v;
}

// ---------------------------------------------------------------------------
// Kernel 2: GEMM. Block = 256 threads (8 waves), one 16-row M tile per block.
// Wave w owns N-tiles [5w, 5w+4]. A chunk staged in LDS, B streamed from L2.
// ---------------------------------------------------------------------------
__global__ __launch_bounds__(256)
void ISTFT_gemm(const float* __restrict__ X, const float* __restrict__ Wf,
                float* __restrict__ out) {
  __shared__ float As[16 * LDSTR];             // 49,408 B

  const int tid  = threadIdx.x;
  const int wave = tid >> 5;
  const int lane = tid & 31;
  const int lrow = lane & 15;
  const int lhi  = lane >> 4;
  const int m0   = blockIdx.x << 4;            // batch-row tile base

  v8f acc[5] = {};                             // five 16x16 f32 accumulators

  const float* __restrict__ Xblk = X + (size_t)m0 * KTOT;
  const float* __restrict__ Arow = As + lrow * LDSTR + (lhi << 1);
  const float* __restrict__ Wb   = Wf + (size_t)(wave * 5) * FRAG + lane * 2;

  for (int kbase = 0; kbase < KTOT; kbase += KCH) {
    const int klen = (KTOT - kbase < KCH) ? (KTOT - kbase) : KCH;  // 768 or 12

    // stage A tile: 16 rows x klen cols, padded LDS stride
    for (int idx = tid; idx < 16 * klen; idx += 256) {
      int rr = idx / klen;
      int cc = idx - rr * klen;
      As[rr * LDSTR + cc] = Xblk[(size_t)rr * KTOT + kbase + cc];
    }
    __syncthreads();

    const int ksteps = klen >> 2;
    const int kk0    = kbase >> 2;
    for (int kk = 0; kk < ksteps; ++kk) {
      v2f a = *(const v2f*)(Arow + (kk << 2));                 // ds_read_b64
      const float* wr = Wb + (size_t)(kk0 + kk) * (NTILES * FRAG);
      #pragma unroll
      for (int t = 0; t < 5; ++t) {
        v2f b = *(const v2f*)(wr + t * FRAG);                  // global b64, coalesced
        acc[t] = __builtin_amdgcn_wmma_f32_16x16x4_f32(
            false, a, false, b, (short)0, acc[t], false, false);
      }
    }
    __syncthreads();
  }

  // store: C/D layout -> row = m0 + v + 8*(lane>=16), col = lane%16 within tile
  #pragma unroll
  for (int t = 0; t < 5; ++t) {
    int colg = (wave * 5 + t) * 16 + lrow;
    if (colg < NOUT) {
      #pragma unroll
      for (int v = 0; v < 8; ++v) {
        int row = m0 + v + (lhi << 3);
        out[(size_t)row * NOUT + colg] = acc[t][v];
      }
    }
  }
}

// ---------------------------------------------------------------------------
extern "C" void kernel_launch(void* const* d_in, const int* in_sizes, int n_in,
                              void* d_out, int out_size, void* d_ws, size_t ws_size,
                              hipStream_t stream) {
  const float* X  = (const float*)d_in[0];     // [8192,1,257,6,2] fp32
  float*       O  = (float*)d_out;             // [8192,1,600] fp32
  float*       Wf = (float*)d_ws;              // 771*40*64 floats ~= 7.5 MB

  const int wtot    = KSTEPS * NTILES * FRAG;  // 1,973,760
  const int wblocks = (wtot + 255) / 256;

  ISTFT_buildW<<<wblocks, 256, 0, stream>>>(Wf);
  ISTFT_gemm<<<8192 / 16, 256, 0, stream>>>(X, Wf, O);
}